// GCN_6562710028851
// MI455X (gfx1250) — compile-verified
//
#include <hip/hip_runtime.h>
#include <hip/hip_bf16.h>

typedef float v2f __attribute__((ext_vector_type(2)));
typedef float v8f __attribute__((ext_vector_type(8)));

#define N_NODES_C   100000
#define D           64
#define NUM_GRAPHS  64
#define BN_EPS      1e-5f
#define LDS_STRIDE  68   // transposed W: bank(4c+k); halves k0/k0+2 -> disjoint bank sets

// ---------------------------------------------------------------------------
// Degree / normalization:  deg[i] = 1 (self loop) + #edges with dst==i
// dinv[i] = rsqrt(deg[i])   (deg >= 1 so max(deg,1) is a no-op)
// ---------------------------------------------------------------------------
__global__ void deg_init_kernel(float* __restrict__ deg, int n) {
    int i = blockIdx.x * blockDim.x + threadIdx.x;
    if (i < n) deg[i] = 1.0f;
}

__global__ void deg_count_kernel(const int* __restrict__ dst, float* __restrict__ deg, int ne) {
    int e = blockIdx.x * blockDim.x + threadIdx.x;
    if (e < ne) atomicAdd(&deg[dst[e]], 1.0f);
}

__global__ void deg_finalize_kernel(float* __restrict__ deg, int n) {
    int i = blockIdx.x * blockDim.x + threadIdx.x;
    if (i < n) deg[i] = rsqrtf(deg[i]);
}

// ---------------------------------------------------------------------------
// GEMM: Y[n x 64] = X[n x 64] @ W[64 x 64] using V_WMMA_F32_16X16X4_F32.
// One wave owns a 16-row tile; 4 accumulator tiles of 16x16 cover N=64.
// W is staged TRANSPOSED in LDS so each B fragment (W[k][c], W[k+1][c]) is a
// single aligned ds_load_b64 into an even VGPR pair (no shuffle moves).
// Block = 320 threads (10 waves) -> 625 blocks * 10 * 16 = 100000 rows exact.
// ---------------------------------------------------------------------------
__global__ void gemm_n64_wmma_kernel(const float* __restrict__ X,
                                     const float* __restrict__ W,
                                     float* __restrict__ Y) {
    __shared__ __align__(16) float WsT[D * LDS_STRIDE];   // WsT[c*68 + k] = W[k][c]
    for (int i = threadIdx.x; i < D * D; i += blockDim.x) {
        int k = i >> 6, c = i & 63;
        WsT[c * LDS_STRIDE + k] = W[i];
    }
    __syncthreads();

    const int wave  = blockIdx.x * (blockDim.x >> 5) + (threadIdx.x >> 5);
    const int lane  = threadIdx.x & 31;
    const int lhalf = lane >> 4;     // 0: K=k0,k0+1 ; 1: K=k0+2,k0+3
    const int lmod  = lane & 15;
    const int row0  = wave * 16;
    const float* __restrict__ Xrow = X + (size_t)(row0 + lmod) * D;

    v8f acc[4] = {};  // four 16x16 f32 accumulator tiles (N = 0..63)

    #pragma unroll
    for (int k0 = 0; k0 < D; k0 += 4) {
        const int ka = k0 + lhalf * 2;
        // A fragment 16x4 (ISA layout): lane<16 -> K=k0,k0+1 ; lane>=16 -> K=k0+2,k0+3
        v2f a = *(const v2f*)(Xrow + ka);
        #pragma unroll
        for (int n = 0; n < 4; ++n) {
            // B fragment 4x16: contiguous (K=ka, K=ka+1) at column n*16+lmod
            v2f b = *(const v2f*)(&WsT[(n * 16 + lmod) * LDS_STRIDE + ka]);
            acc[n] = __builtin_amdgcn_wmma_f32_16x16x4_f32(
                false, a, false, b, (short)0, acc[n], false, false);
        }
    }

    // D tile layout: VGPR j -> row row0 + j + 8*lhalf, col n*16 + lmod
    #pragma unroll
    for (int n = 0; n < 4; ++n) {
        #pragma unroll
        for (int j = 0; j < 8; ++j) {
            int r = row0 + j + lhalf * 8;
            Y[(size_t)r * D + n * 16 + lmod] = acc[n][j];
        }
    }
}

// ---------------------------------------------------------------------------
// Self-loop contribution: out[i,:] = h[i,:] * dinv[i]^2
// ---------------------------------------------------------------------------
__global__ void self_loop_init_kernel(const float* __restrict__ H,
                                      const float* __restrict__ dinv,
                                      float* __restrict__ out, int n) {
    size_t t = (size_t)blockIdx.x * blockDim.x + threadIdx.x;
    if (t >= (size_t)n * D) return;
    int node = (int)(t >> 6);
    float w = dinv[node];
    out[t] = H[t] * (w * w);
}

// ---------------------------------------------------------------------------
// Edge scatter: out[dst,:] += H[src,:] * dinv[src]*dinv[dst]
// 16 threads per edge, float4 gather (L2-resident), f32 atomics into L2.
// ---------------------------------------------------------------------------
__global__ void edge_scatter_kernel(const float* __restrict__ H,
                                    const int* __restrict__ src,
                                    const int* __restrict__ dst,
                                    const float* __restrict__ dinv,
                                    float* __restrict__ out, int ne) {
    long long t = (long long)blockIdx.x * blockDim.x + threadIdx.x;
    int e = (int)(t >> 4);
    if (e >= ne) return;
    int c = (int)(t & 15) * 4;
    int s = src[e], d = dst[e];
    float w = dinv[s] * dinv[d];
    float4 hv = *(const float4*)(H + (size_t)s * D + c);
    float* o = out + (size_t)d * D + c;
    atomicAdd(o + 0, hv.x * w);
    atomicAdd(o + 1, hv.y * w);
    atomicAdd(o + 2, hv.z * w);
    atomicAdd(o + 3, hv.w * w);
}

// ---------------------------------------------------------------------------
// Epilogue: y = relu( (agg + bias - mean) * (gamma * rsqrt(var+eps)) + beta )
// ---------------------------------------------------------------------------
__global__ void bias_bn_relu_kernel(const float* __restrict__ agg,
                                    const float* __restrict__ bias,
                                    const float* __restrict__ g,
                                    const float* __restrict__ be,
                                    const float* __restrict__ m,
                                    const float* __restrict__ v,
                                    float* __restrict__ out, int n) {
    size_t t = (size_t)blockIdx.x * blockDim.x + threadIdx.x;
    if (t >= (size_t)n * D) return;
    int c = (int)(t & 63);
    float scale = g[c] * rsqrtf(v[c] + BN_EPS);
    float y = (agg[t] + bias[c] - m[c]) * scale + be[c];
    out[t] = fmaxf(y, 0.0f);
}

// ---------------------------------------------------------------------------
// Pooling: batch is sorted, so each thread (one channel, 64-node run)
// accumulates locally and emits an atomic only on graph boundary.
// ---------------------------------------------------------------------------
__global__ void zero_kernel(float* __restrict__ p, int n) {
    int i = blockIdx.x * blockDim.x + threadIdx.x;
    if (i < n) p[i] = 0.0f;
}

__global__ void pool_accum_kernel(const float* __restrict__ H,
                                  const int* __restrict__ batch,
                                  float* __restrict__ sums,
                                  float* __restrict__ cnt, int n) {
    int c     = threadIdx.x & 63;
    int chunk = blockIdx.x * (blockDim.x >> 6) + (threadIdx.x >> 6);
    int n0 = chunk * 64;
    if (n0 >= n) return;
    int n1 = n0 + 64; if (n1 > n) n1 = n;
    int curb = batch[n0];
    float acc = 0.0f, cacc = 0.0f;
    for (int i = n0; i < n1; ++i) {
        int b = batch[i];
        if (b != curb) {
            atomicAdd(&sums[(size_t)curb * D + c], acc);
            if (c == 0) atomicAdd(&cnt[curb], cacc);
            curb = b; acc = 0.0f; cacc = 0.0f;
        }
        acc  += H[(size_t)i * D + c];
        cacc += 1.0f;
    }
    atomicAdd(&sums[(size_t)curb * D + c], acc);
    if (c == 0) atomicAdd(&cnt[curb], cacc);
}

// Final classifier: out[g, o] = (sums[g,:]/max(cnt,1)) @ Wc[:,o] + bc[o]
__global__ void classify_kernel(const float* __restrict__ sums,
                                const float* __restrict__ cnt,
                                const float* __restrict__ Wc,
                                const float* __restrict__ bc,
                                float* __restrict__ out) {
    int t = threadIdx.x;           // 128 threads: 64 graphs x 2 outputs
    if (t >= NUM_GRAPHS * 2) return;
    int g = t >> 1, o = t & 1;
    float inv = 1.0f / fmaxf(cnt[g], 1.0f);
    float acc = bc[o];
    #pragma unroll 8
    for (int c = 0; c < D; ++c)
        acc += sums[(size_t)g * D + c] * inv * Wc[c * 2 + o];
    out[g * 2 + o] = acc;
}

// ---------------------------------------------------------------------------
extern "C" void kernel_launch(void* const* d_in, const int* in_sizes, int n_in,
                              void* d_out, int out_size, void* d_ws, size_t ws_size,
                              hipStream_t stream) {
    const float* x    = (const float*)d_in[0];
    const int*   ei   = (const int*)d_in[1];
    const int*   batch= (const int*)d_in[2];
    const float* W1   = (const float*)d_in[3];
    const float* b1   = (const float*)d_in[4];
    const float* g1   = (const float*)d_in[5];
    const float* be1  = (const float*)d_in[6];
    const float* m1   = (const float*)d_in[7];
    const float* v1   = (const float*)d_in[8];
    const float* W2   = (const float*)d_in[9];
    const float* b2   = (const float*)d_in[10];
    const float* g2   = (const float*)d_in[11];
    const float* be2  = (const float*)d_in[12];
    const float* m2   = (const float*)d_in[13];
    const float* v2   = (const float*)d_in[14];
    const float* Wc   = (const float*)d_in[15];
    const float* bc   = (const float*)d_in[16];

    const int n  = in_sizes[0] / D;        // 100000 nodes
    const int ne = in_sizes[1] / 2;        // 1200000 edges
    const int*   src = ei;
    const int*   dst = ei + ne;

    // Workspace layout (floats)
    float* bufA = (float*)d_ws;                       // n*64
    float* bufB = bufA + (size_t)n * D;               // n*64
    float* dinv = bufB + (size_t)n * D;               // n
    float* sums = dinv + n;                           // 64*64
    float* cnt  = sums + (size_t)NUM_GRAPHS * D;      // 64

    const size_t nelem = (size_t)n * D;
    const int B256 = 256;
    const int gNode  = (n + B256 - 1) / B256;
    const int gEdge  = (ne + B256 - 1) / B256;
    const int gElem  = (int)((nelem + B256 - 1) / B256);
    const int gScat  = (int)(((long long)ne * 16 + B256 - 1) / B256);
    const int gGemm  = (n / 16) / 10;                 // 625 blocks of 10 waves
    const int gPool  = ((n + 63) / 64 + 3) / 4;       // 4 chunks per 256-thread block

    // --- normalization coefficients -------------------------------------
    deg_init_kernel    <<<gNode, B256, 0, stream>>>(dinv, n);
    deg_count_kernel   <<<gEdge, B256, 0, stream>>>(dst, dinv, ne);
    deg_finalize_kernel<<<gNode, B256, 0, stream>>>(dinv, n);

    // --- layer 1 ---------------------------------------------------------
    gemm_n64_wmma_kernel <<<gGemm, 320, 0, stream>>>(x, W1, bufA);
    self_loop_init_kernel<<<gElem, B256, 0, stream>>>(bufA, dinv, bufB, n);
    edge_scatter_kernel  <<<gScat, B256, 0, stream>>>(bufA, src, dst, dinv, bufB, ne);
    bias_bn_relu_kernel  <<<gElem, B256, 0, stream>>>(bufB, b1, g1, be1, m1, v1, bufA, n);

    // --- layer 2 ---------------------------------------------------------
    gemm_n64_wmma_kernel <<<gGemm, 320, 0, stream>>>(bufA, W2, bufB);
    self_loop_init_kernel<<<gElem, B256, 0, stream>>>(bufB, dinv, bufA, n);
    edge_scatter_kernel  <<<gScat, B256, 0, stream>>>(bufB, src, dst, dinv, bufA, ne);
    bias_bn_relu_kernel  <<<gElem, B256, 0, stream>>>(bufA, b2, g2, be2, m2, v2, bufB, n);

    // --- pool + classify --------------------------------------------------
    zero_kernel      <<<(NUM_GRAPHS * D + NUM_GRAPHS + B256 - 1) / B256, B256, 0, stream>>>(sums, NUM_GRAPHS * D + NUM_GRAPHS);
    pool_accum_kernel<<<gPool, B256, 0, stream>>>(bufB, batch, sums, cnt, n);
    classify_kernel  <<<1, 128, 0, stream>>>(sums, cnt, Wc, bc, (float*)d_out);
}